// CompressModel_47038481826436
// MI455X (gfx1250) — compile-verified
//
#include <hip/hip_runtime.h>
#include <hip/hip_bf16.h>
#include <math.h>

#define DIM 1024
#define EPS 1e-5f
#define RS 1028                    // padded LDS row stride in floats (16B aligned, bank-spread)
#define ROWS_PER_WAVE 16
#define WAVES_PER_BLOCK 2
#define BLOCK_THREADS (WAVES_PER_BLOCK * 32)
#define ROWS_TOTAL (8 * 4096)
#define N_ELEMS ((long long)ROWS_TOTAL * DIM)

typedef __attribute__((ext_vector_type(2))) float v2f;
typedef __attribute__((ext_vector_type(8))) float v8f;
typedef int v4i __attribute__((vector_size(16)));  // matches builtin's expected pointee

#if __has_builtin(__builtin_amdgcn_wmma_f32_16x16x4_f32)
#define HAVE_WMMA_F32 1
#endif

#if __has_builtin(__builtin_amdgcn_global_load_async_to_lds_b128) && \
    __has_builtin(__builtin_amdgcn_s_wait_asynccnt)
#define HAVE_ASYNC_LDS 1
typedef __attribute__((address_space(1))) v4i* glob_b128_t;
typedef __attribute__((address_space(3))) v4i* lds_b128_t;
#endif

__device__ __forceinline__ float had16(int i, int j) {
  return (__builtin_popcount(i & j) & 1) ? -1.0f : 1.0f;
}

// Apply H16 along one digit of the column index, batched over the wave's 16 rows.
// Column for digit-value k (with "other digits" offset o) is: o + k*stride.
// stride=4,  omul=64  -> middle digit b (o = a*64 + c)
// stride=64, omul=4   -> high   digit a (o = b*4  + c)
#if HAVE_WMMA_F32
__device__ __forceinline__ void h16_stage(float* __restrict__ T, int lane,
                                          int stride, int omul, const v2f* B) {
  const int half = lane >> 4;      // which K-half of the A/B operands this lane holds
  const int r = lane & 15;         // A-operand row (M index)
  const int n = lane & 15;         // D-operand column (N index)
  for (int i = 0; i < 64; ++i) {
    const int o = (i >> 2) * omul + (i & 3);
    v8f acc = {0.f, 0.f, 0.f, 0.f, 0.f, 0.f, 0.f, 0.f};
#pragma unroll
    for (int s = 0; s < 4; ++s) {  // K=16 as 4 chained K=4 WMMAs
      const int kb = 4 * s + 2 * half;
      v2f a;
      a.x = T[r * RS + o + kb * stride];
      a.y = T[r * RS + o + (kb + 1) * stride];
      acc = __builtin_amdgcn_wmma_f32_16x16x4_f32(false, a, false, B[s],
                                                  (short)0, acc, false, false);
    }
#pragma unroll
    for (int v = 0; v < 8; ++v)    // D: VGPR v holds rows v (lanes 0-15) / v+8 (lanes 16-31)
      T[(v + 8 * half) * RS + o + n * stride] = acc[v];
  }
}
#else
// Scalar fallback: same transform via register butterflies.
__device__ __forceinline__ void h16_stage(float* __restrict__ T, int lane,
                                          int stride, int omul) {
  for (int i = 0; i < 32; ++i) {
    const int t = lane + 32 * i;   // 16 rows x 64 offsets = 1024 tasks
    const int r = t >> 6;
    const int oi = t & 63;
    const int o = (oi >> 2) * omul + (oi & 3);
    float v[16];
#pragma unroll
    for (int k = 0; k < 16; ++k) v[k] = T[r * RS + o + k * stride];
#pragma unroll
    for (int h = 1; h < 16; h <<= 1)
#pragma unroll
      for (int k = 0; k < 16; ++k)
        if ((k & h) == 0) {
          float p = v[k], q = v[k + h];
          v[k] = p + q;
          v[k + h] = p - q;
        }
#pragma unroll
    for (int k = 0; k < 16; ++k) T[r * RS + o + k * stride] = v[k];
  }
}
#endif

__global__ void fwht_pass1(const float* __restrict__ x,
                           const float* __restrict__ golay,
                           float* __restrict__ out,
                           double* __restrict__ sumsq) {
  extern __shared__ float smem[];
  const int lane = threadIdx.x & 31;
  const int wave = threadIdx.x >> 5;
  float* T = smem + wave * (ROWS_PER_WAVE * RS);

  const long long tile = (long long)blockIdx.x * WAVES_PER_BLOCK + wave;
  const long long row0 = tile * ROWS_PER_WAVE;
  const float* src = x + row0 * DIM;
  float* dst = out + row0 * DIM;

  // ---- stage the 16x1024 fp32 tile into LDS (async DMA path if available) ----
#if HAVE_ASYNC_LDS
#pragma unroll
  for (int r = 0; r < ROWS_PER_WAVE; ++r) {
#pragma unroll
    for (int j = 0; j < 8; ++j) {
      const int e = (lane + 32 * j) * 4;
      __builtin_amdgcn_global_load_async_to_lds_b128(
          (glob_b128_t)(src + r * DIM + e),
          (lds_b128_t)(T + r * RS + e), 0, 0);
    }
  }
#else
  for (int r = 0; r < ROWS_PER_WAVE; ++r) {
#pragma unroll
    for (int j = 0; j < 8; ++j) {
      const int e = (lane + 32 * j) * 4;
      float4 v = *(const float4*)(src + r * DIM + e);
      *(float4*)(T + r * RS + e) = v;
    }
  }
#endif

  // Preload the Golay float4s this lane needs (hides async latency).
  float4 gv[8];
  const float4* g4 = (const float4*)golay;
#pragma unroll
  for (int j = 0; j < 8; ++j) gv[j] = g4[lane + 32 * j];

#if HAVE_ASYNC_LDS
  __builtin_amdgcn_s_wait_asynccnt(0);
#endif

  // ---- stage 1: Golay modulation fused with H4 on the low digit (contiguous 4) ----
  for (int r = 0; r < ROWS_PER_WAVE; ++r) {
#pragma unroll
    for (int j = 0; j < 8; ++j) {
      float* p = T + r * RS + (lane + 32 * j) * 4;
      float4 v = *(float4*)p;
      const float a = v.x * gv[j].x, b = v.y * gv[j].y;
      const float c = v.z * gv[j].z, d = v.w * gv[j].w;
      const float s0 = a + b, d0 = a - b, s1 = c + d, d1 = c - d;
      v.x = s0 + s1;
      v.y = d0 + d1;
      v.z = s0 - s1;
      v.w = d0 - d1;
      *(float4*)p = v;
    }
  }

  // ---- stages 2 & 3: H16 on middle digit (stride 4) and high digit (stride 64) ----
#if HAVE_WMMA_F32
  v2f B[4];
  {
    const int half = lane >> 4, n = lane & 15;
#pragma unroll
    for (int s = 0; s < 4; ++s) {
      const int kb = 4 * s + 2 * half;
      B[s].x = had16(kb, n);
      B[s].y = had16(kb + 1, n);
    }
  }
  h16_stage(T, lane, 4, 64, B);
  h16_stage(T, lane, 64, 4, B);
#else
  h16_stage(T, lane, 4, 64);
  h16_stage(T, lane, 64, 4);
#endif

  // ---- write back transformed tile and accumulate sum of squares ----
  float ssq = 0.f;
  for (int r = 0; r < ROWS_PER_WAVE; ++r) {
#pragma unroll
    for (int j = 0; j < 8; ++j) {
      const int e = (lane + 32 * j) * 4;
      float4 v = *(const float4*)(T + r * RS + e);
      ssq += v.x * v.x + v.y * v.y + v.z * v.z + v.w * v.w;
      *(float4*)(dst + r * DIM + e) = v;
    }
  }
#pragma unroll
  for (int m = 16; m >= 1; m >>= 1) ssq += __shfl_xor(ssq, m, 32);
  if (lane == 0) atomicAdd(sumsq, (double)ssq);
}

__global__ void scale_pass2(float* __restrict__ out,
                            const double* __restrict__ sumsq) {
  const double mean = *sumsq / (double)N_ELEMS;
  const float alpha = (float)sqrt(mean + (double)EPS);
  const float s = 1.0f / (alpha + EPS);
  long long i = ((long long)blockIdx.x * blockDim.x + threadIdx.x) * 4;
  const long long stride = (long long)gridDim.x * blockDim.x * 4;
  for (; i < N_ELEMS; i += stride) {
    float4 v = *(float4*)(out + i);
    v.x *= s;
    v.y *= s;
    v.z *= s;
    v.w *= s;
    *(float4*)(out + i) = v;
  }
}

extern "C" void kernel_launch(void* const* d_in, const int* in_sizes, int n_in,
                              void* d_out, int out_size, void* d_ws, size_t ws_size,
                              hipStream_t stream) {
  (void)in_sizes;
  (void)n_in;
  (void)out_size;
  (void)ws_size;
  const float* x = (const float*)d_in[0];
  const float* golay = (const float*)d_in[1];
  float* out = (float*)d_out;
  double* ss = (double*)d_ws;

  const size_t smem_bytes =
      (size_t)WAVES_PER_BLOCK * ROWS_PER_WAVE * RS * sizeof(float);  // 131584 B
  (void)hipFuncSetAttribute((const void*)fwht_pass1,
                            hipFuncAttributeMaxDynamicSharedMemorySize,
                            (int)smem_bytes);

  (void)hipMemsetAsync(d_ws, 0, sizeof(double), stream);

  const int grid1 = ROWS_TOTAL / (ROWS_PER_WAVE * WAVES_PER_BLOCK);  // 1024
  fwht_pass1<<<grid1, BLOCK_THREADS, smem_bytes, stream>>>(x, golay, out, ss);
  scale_pass2<<<2048, 256, 0, stream>>>(out, ss);
}